// Plane_refine_block_4_41927470743689
// MI455X (gfx1250) — compile-verified
//
#include <hip/hip_runtime.h>

typedef __attribute__((ext_vector_type(2))) float v2f;
typedef __attribute__((ext_vector_type(8))) float v8f;

#define WPB 2              // waves per block in phase1
#define H1STRIDE 36        // padded LDS row stride (floats)

__device__ __forceinline__ void lds_fence() {
  asm volatile("s_wait_dscnt 0" ::: "memory");
}

__device__ __forceinline__ void atomic_max_f(float* a, float v) {
  if (v >= 0.0f) atomicMax((int*)a, __float_as_int(v));
  else           atomicMin((unsigned int*)a, __float_as_uint(v));
}

// ---------------------------------------------------------------- init
__global__ void k_init(unsigned* cnt_glob, float* mp_glob) {
  int i = blockIdx.x * blockDim.x + threadIdx.x;
  if (i < 32)   cnt_glob[i] = 0u;
  if (i < 1024) mp_glob[i] = __int_as_float(0xff800000); // -inf
}

// ---------------------------------------------------------------- phase 1
// Per 32-column tile (2x 16-col WMMA tiles): h1 = relu(W1@X+b1),
// h = relu(W2@h1+b2), att = Watt@h ; store h,att to ws ; mask bits + mp max.
__global__ __launch_bounds__(WPB * 32) void k_phase1(
    const float* __restrict__ xyz, const float* __restrict__ center,
    const float* __restrict__ feature, const float* __restrict__ feature_geo,
    const float* __restrict__ plane_centers, const float* __restrict__ normals,
    const float* __restrict__ xyz_min, const float* __restrict__ xyz_max,
    const int* __restrict__ axes,
    const float* __restrict__ W1, const float* __restrict__ b1,
    const float* __restrict__ W2, const float* __restrict__ b2,
    const float* __restrict__ Watt,
    float* __restrict__ h_ws, float* __restrict__ att_ws,
    unsigned* __restrict__ maskbits_ws, unsigned* __restrict__ cnt_glob,
    float* __restrict__ mp_glob, int N) {
  __shared__ float sW1[64 * 64];
  __shared__ float sW2[32 * 64];
  __shared__ float sWatt[32 * 32];
  __shared__ float sB1[64], sB2[32];
  __shared__ float sNrm[96], sDp[32], sMn[96], sMx[96];
  __shared__ int   sAx[64];
  __shared__ float sMp[1024];
  __shared__ unsigned sCnt[32];
  __shared__ float sH1[WPB][64 * H1STRIDE];
  __shared__ float sH[WPB][32 * H1STRIDE];

  const int tid = threadIdx.x;
  const int lane = tid & 31;
  const int w = tid >> 5;

  for (int i = tid; i < 64 * 64; i += blockDim.x) sW1[i] = W1[i];
  for (int i = tid; i < 32 * 64; i += blockDim.x) sW2[i] = W2[i];
  for (int i = tid; i < 32 * 32; i += blockDim.x) sWatt[i] = Watt[i];
  for (int i = tid; i < 64; i += blockDim.x) { sB1[i] = b1[i]; sAx[i] = axes[i]; }
  for (int i = tid; i < 32; i += blockDim.x) sB2[i] = b2[i];
  for (int i = tid; i < 96; i += blockDim.x) {
    sNrm[i] = normals[i]; sMn[i] = xyz_min[i]; sMx[i] = xyz_max[i];
  }
  for (int i = tid; i < 1024; i += blockDim.x) sMp[i] = __int_as_float(0xff800000);
  if (tid < 32) {
    sDp[tid] = -(plane_centers[tid * 3 + 0] * normals[tid * 3 + 0] +
                 plane_centers[tid * 3 + 1] * normals[tid * 3 + 1] +
                 plane_centers[tid * 3 + 2] * normals[tid * 3 + 2]);
    sCnt[tid] = 0u;
  }
  __syncthreads();

  const float cx = center[0], cy = center[1], cz = center[2];
  const int ntiles = (N + 31) / 32;
  const int lcol = lane & 15;
  const int hi = lane >> 4;   // 0 for lanes 0-15, 1 for lanes 16-31
  float* h1l = sH1[w];
  float* hl = sH[w];

  for (int tile = blockIdx.x * WPB + w; tile < ntiles; tile += gridDim.x * WPB) {
    const int n0 = tile * 32;
    int colA = n0 + lcol;        if (colA > N - 1) colA = N - 1;
    int colB = n0 + 16 + lcol;   if (colB > N - 1) colB = N - 1;

    // ---- layer 1: 64x64 @ 64x32
    v8f acc1[2][4];
    #pragma unroll
    for (int t = 0; t < 2; ++t)
      #pragma unroll
      for (int mt = 0; mt < 4; ++mt) acc1[t][mt] = (v8f){};
    for (int kc = 0; kc < 16; ++kc) {
      const int krow = kc * 4 + hi * 2;
      const float* xb = (krow < 32 ? feature : feature_geo) + (size_t)(krow & 31) * N;
      v2f bA, bB;
      bA.x = xb[colA]; bA.y = xb[(size_t)N + colA];
      bB.x = xb[colB]; bB.y = xb[(size_t)N + colB];
      #pragma unroll
      for (int mt = 0; mt < 4; ++mt) {
        const float* wp = &sW1[(mt * 16 + lcol) * 64 + kc * 4 + hi * 2];
        v2f a; a.x = wp[0]; a.y = wp[1];
        acc1[0][mt] = __builtin_amdgcn_wmma_f32_16x16x4_f32(false, a, false, bA, (short)0, acc1[0][mt], false, false);
        acc1[1][mt] = __builtin_amdgcn_wmma_f32_16x16x4_f32(false, a, false, bB, (short)0, acc1[1][mt], false, false);
      }
    }
    // bias + relu, stage D-layout -> LDS (row-major 64 x 32)
    #pragma unroll
    for (int t = 0; t < 2; ++t)
      #pragma unroll
      for (int mt = 0; mt < 4; ++mt)
        #pragma unroll
        for (int r = 0; r < 8; ++r) {
          const int row = mt * 16 + r + hi * 8;
          h1l[row * H1STRIDE + t * 16 + lcol] = fmaxf(acc1[t][mt][r] + sB1[row], 0.0f);
        }
    lds_fence();

    // ---- layer 2: 32x64 @ 64x32
    v8f acc2[2][2];
    #pragma unroll
    for (int t = 0; t < 2; ++t)
      #pragma unroll
      for (int mt = 0; mt < 2; ++mt) acc2[t][mt] = (v8f){};
    for (int kc = 0; kc < 16; ++kc) {
      const int krow = kc * 4 + hi * 2;
      v2f bA, bB;
      bA.x = h1l[krow * H1STRIDE + lcol];        bA.y = h1l[(krow + 1) * H1STRIDE + lcol];
      bB.x = h1l[krow * H1STRIDE + 16 + lcol];   bB.y = h1l[(krow + 1) * H1STRIDE + 16 + lcol];
      #pragma unroll
      for (int mt = 0; mt < 2; ++mt) {
        const float* wp = &sW2[(mt * 16 + lcol) * 64 + kc * 4 + hi * 2];
        v2f a; a.x = wp[0]; a.y = wp[1];
        acc2[0][mt] = __builtin_amdgcn_wmma_f32_16x16x4_f32(false, a, false, bA, (short)0, acc2[0][mt], false, false);
        acc2[1][mt] = __builtin_amdgcn_wmma_f32_16x16x4_f32(false, a, false, bB, (short)0, acc2[1][mt], false, false);
      }
    }
    // bias + relu -> h : stage to LDS + store to ws
    #pragma unroll
    for (int t = 0; t < 2; ++t)
      #pragma unroll
      for (int mt = 0; mt < 2; ++mt)
        #pragma unroll
        for (int r = 0; r < 8; ++r) {
          const int row = mt * 16 + r + hi * 8;
          const float v = fmaxf(acc2[t][mt][r] + sB2[row], 0.0f);
          hl[row * H1STRIDE + t * 16 + lcol] = v;
          const int n = n0 + t * 16 + lcol;
          if (n < N) h_ws[(size_t)row * N + n] = v;
        }
    lds_fence();

    // ---- att = Watt @ h : 32x32 @ 32x32
    v8f acc3[2][2];
    #pragma unroll
    for (int t = 0; t < 2; ++t)
      #pragma unroll
      for (int mt = 0; mt < 2; ++mt) acc3[t][mt] = (v8f){};
    for (int kc = 0; kc < 8; ++kc) {
      const int krow = kc * 4 + hi * 2;
      v2f bA, bB;
      bA.x = hl[krow * H1STRIDE + lcol];        bA.y = hl[(krow + 1) * H1STRIDE + lcol];
      bB.x = hl[krow * H1STRIDE + 16 + lcol];   bB.y = hl[(krow + 1) * H1STRIDE + 16 + lcol];
      #pragma unroll
      for (int mt = 0; mt < 2; ++mt) {
        const float* wp = &sWatt[(mt * 16 + lcol) * 32 + kc * 4 + hi * 2];
        v2f a; a.x = wp[0]; a.y = wp[1];
        acc3[0][mt] = __builtin_amdgcn_wmma_f32_16x16x4_f32(false, a, false, bA, (short)0, acc3[0][mt], false, false);
        acc3[1][mt] = __builtin_amdgcn_wmma_f32_16x16x4_f32(false, a, false, bB, (short)0, acc3[1][mt], false, false);
      }
    }
    #pragma unroll
    for (int t = 0; t < 2; ++t)
      #pragma unroll
      for (int mt = 0; mt < 2; ++mt)
        #pragma unroll
        for (int r = 0; r < 8; ++r) {
          const int row = mt * 16 + r + hi * 8;
          const int n = n0 + t * 16 + lcol;
          if (n < N) att_ws[(size_t)row * N + n] = acc3[t][mt][r];
        }

    // ---- per-point: mask bits + mp max reduction (1 point per lane)
    const int n = n0 + lane;
    if (n < N) {
      const float px = xyz[n * 3 + 0] + cx;
      const float py = xyz[n * 3 + 1] + cy;
      const float pz = xyz[n * 3 + 2] + cz;
      unsigned mb = 0u;
      for (int p = 0; p < 32; ++p) {
        const float s = px * sNrm[p * 3 + 0] + py * sNrm[p * 3 + 1] + pz * sNrm[p * 3 + 2] + sDp[p];
        const int a0 = sAx[p * 2 + 0], a1 = sAx[p * 2 + 1];
        const float v0 = (a0 == 0) ? px : ((a0 == 1) ? py : pz);
        const float v1 = (a1 == 0) ? px : ((a1 == 1) ? py : pz);
        const bool rm = (v0 >= sMn[p * 3 + a0]) && (v0 < sMx[p * 3 + a0]) &&
                        (v1 >= sMn[p * 3 + a1]) && (v1 < sMx[p * 3 + a1]);
        if (fabsf(s) < 0.1f && rm) mb |= (1u << p);
      }
      maskbits_ws[n] = mb;
      if (mb) {
        float hreg[32];
        #pragma unroll
        for (int d = 0; d < 32; ++d) hreg[d] = hl[d * H1STRIDE + lane];
        unsigned m = mb;
        while (m) {
          const int p = __ffs(m) - 1; m &= m - 1;
          atomicAdd(&sCnt[p], 1u);
          #pragma unroll
          for (int d = 0; d < 32; ++d)
            atomicMax((int*)&sMp[p * 32 + d], __float_as_int(hreg[d])); // hreg >= 0
        }
      }
    }
  }

  __syncthreads();
  for (int i = tid; i < 1024; i += blockDim.x) atomic_max_f(&mp_glob[i], sMp[i]);
  if (tid < 32) atomicAdd(&cnt_glob[tid], sCnt[tid]);
}

// ---------------------------------------------------------------- plane MLP
__global__ __launch_bounds__(256) void k_planes(
    const float* __restrict__ pc, const float* __restrict__ nrm,
    const float* __restrict__ mn, const float* __restrict__ mx,
    const float* __restrict__ W3, const float* __restrict__ b3,
    const float* __restrict__ W4, const float* __restrict__ b4,
    const unsigned* __restrict__ cnt_glob, const float* __restrict__ mp_glob,
    float* __restrict__ np_ws, unsigned* __restrict__ vb_ws,
    float* mx_on, float* mx_off, float* den_on, float* den_off,
    float* agg_on, float* agg_off, float* __restrict__ out, int N) {
  __shared__ float sMp[1024];
  __shared__ unsigned sValid;
  const int tid = threadIdx.x;
  if (tid == 0) sValid = 0u;
  __syncthreads();
  if (tid < 32 && cnt_glob[tid] > 1u) atomicOr(&sValid, 1u << tid);
  __syncthreads();
  const unsigned vb = sValid;
  for (int i = tid; i < 1024; i += blockDim.x)
    sMp[i] = ((vb >> (i >> 5)) & 1u) ? mp_glob[i] : 0.0f;
  __syncthreads();

  if (tid < 32) {
    const int p = tid;
    float h3[32];
    #pragma unroll 4
    for (int i = 0; i < 32; ++i) {
      float s = b3[i];
      for (int d = 0; d < 32; ++d) s += sMp[p * 32 + d] * W3[i * 32 + d];
      h3[i] = fmaxf(s, 0.0f);
    }
    const float Dp = -(pc[p * 3 + 0] * nrm[p * 3 + 0] + pc[p * 3 + 1] * nrm[p * 3 + 1] +
                       pc[p * 3 + 2] * nrm[p * 3 + 2]);
    float npv[4];
    #pragma unroll
    for (int j = 0; j < 4; ++j) {
      float s = b4[j];
      for (int i = 0; i < 32; ++i) s += h3[i] * W4[j * 32 + i];
      s = fmaxf(s, 0.0f);
      npv[j] = s + ((j < 3) ? nrm[p * 3 + j] : Dp);
      np_ws[p * 4 + j] = npv[j];
    }
    const size_t onOff = (size_t)N * 32 + (size_t)p * 42 + 32;
    const size_t offOff = (size_t)N * 32 + 1344 + (size_t)p * 42 + 32;
    float ori[10] = {npv[0], npv[1], npv[2], npv[3],
                     mn[p * 3], mn[p * 3 + 1], mn[p * 3 + 2],
                     mx[p * 3], mx[p * 3 + 1], mx[p * 3 + 2]};
    for (int q = 0; q < 10; ++q) { out[onOff + q] = ori[q]; out[offOff + q] = ori[q]; }
    if (p == 0) *vb_ws = vb;
  }
  for (int i = tid; i < 1024; i += blockDim.x) {
    mx_on[i] = __int_as_float(0xff800000);
    mx_off[i] = __int_as_float(0xff800000);
    den_on[i] = 0.0f; den_off[i] = 0.0f; agg_on[i] = 0.0f; agg_off[i] = 0.0f;
  }
}

// ---------------------------------------------------------------- projection + on/off + att max
__global__ __launch_bounds__(256) void k_project(
    const float* __restrict__ xyz, const float* __restrict__ center,
    const float* __restrict__ np_ws, const unsigned* __restrict__ vb_ws,
    const unsigned* __restrict__ maskbits_ws, const float* __restrict__ att_ws,
    unsigned* __restrict__ onb, unsigned* __restrict__ offb,
    float* __restrict__ mx_on, float* __restrict__ mx_off,
    float* __restrict__ out, int N) {
  __shared__ float sNp[128];
  __shared__ float sMxOn[1024], sMxOff[1024];
  const int tid = threadIdx.x;
  if (tid < 128) sNp[tid] = np_ws[tid];
  for (int i = tid; i < 1024; i += blockDim.x) {
    sMxOn[i] = __int_as_float(0xff800000);
    sMxOff[i] = __int_as_float(0xff800000);
  }
  __syncthreads();
  const int n = blockIdx.x * blockDim.x + tid;
  if (n < N) {
    const float px = xyz[n * 3 + 0] + center[0];
    const float py = xyz[n * 3 + 1] + center[1];
    const float pz = xyz[n * 3 + 2] + center[2];
    const unsigned m = maskbits_ws[n] & (*vb_ws);
    const size_t maskOff = (size_t)N * 32 + 2688;
    unsigned on = 0u, off = 0u;
    for (int p = 0; p < 32; ++p) {
      const float r = px * sNp[p * 4 + 0] + py * sNp[p * 4 + 1] + pz * sNp[p * 4 + 2] + sNp[p * 4 + 3];
      out[(size_t)n * 32 + p] = r;
      const unsigned bit = (m >> p) & 1u;
      out[maskOff + (size_t)p * N + n] = (float)bit;
      if (bit) { if (r > 0.0f) on |= (1u << p); else off |= (1u << p); }
    }
    onb[n] = on; offb[n] = off;
    if (on | off) {
      float areg[32];
      #pragma unroll
      for (int d = 0; d < 32; ++d) areg[d] = att_ws[(size_t)d * N + n];
      unsigned t = on;
      while (t) {
        const int p = __ffs(t) - 1; t &= t - 1;
        #pragma unroll
        for (int d = 0; d < 32; ++d) atomic_max_f(&sMxOn[p * 32 + d], areg[d]);
      }
      t = off;
      while (t) {
        const int p = __ffs(t) - 1; t &= t - 1;
        #pragma unroll
        for (int d = 0; d < 32; ++d) atomic_max_f(&sMxOff[p * 32 + d], areg[d]);
      }
    }
  }
  __syncthreads();
  for (int i = tid; i < 1024; i += blockDim.x) {
    atomic_max_f(&mx_on[i], sMxOn[i]);
    atomic_max_f(&mx_off[i], sMxOff[i]);
  }
}

// ---------------------------------------------------------------- exp-sum pooling
__global__ __launch_bounds__(256) void k_pool(
    const unsigned* __restrict__ onb, const unsigned* __restrict__ offb,
    const float* __restrict__ att_ws, const float* __restrict__ h_ws,
    const float* __restrict__ mx_on, const float* __restrict__ mx_off,
    float* __restrict__ den_on, float* __restrict__ den_off,
    float* __restrict__ agg_on, float* __restrict__ agg_off, int N) {
  __shared__ float sMxOn[1024], sMxOff[1024];
  __shared__ float sDenOn[1024], sDenOff[1024], sAggOn[1024], sAggOff[1024];
  const int tid = threadIdx.x;
  for (int i = tid; i < 1024; i += blockDim.x) {
    sMxOn[i] = fmaxf(mx_on[i], -10000.0f);
    sMxOff[i] = fmaxf(mx_off[i], -10000.0f);
    sDenOn[i] = 0.0f; sDenOff[i] = 0.0f; sAggOn[i] = 0.0f; sAggOff[i] = 0.0f;
  }
  __syncthreads();
  const int n = blockIdx.x * blockDim.x + tid;
  if (n < N) {
    const unsigned on = onb[n], off = offb[n];
    if (on | off) {
      float areg[32], hreg[32];
      #pragma unroll
      for (int d = 0; d < 32; ++d) {
        areg[d] = att_ws[(size_t)d * N + n];
        hreg[d] = h_ws[(size_t)d * N + n];
      }
      unsigned t = on;
      while (t) {
        const int p = __ffs(t) - 1; t &= t - 1;
        #pragma unroll
        for (int d = 0; d < 32; ++d) {
          const float e = __expf(areg[d] - sMxOn[p * 32 + d]);
          atomicAdd(&sDenOn[p * 32 + d], e);
          atomicAdd(&sAggOn[p * 32 + d], e * hreg[d]);
        }
      }
      t = off;
      while (t) {
        const int p = __ffs(t) - 1; t &= t - 1;
        #pragma unroll
        for (int d = 0; d < 32; ++d) {
          const float e = __expf(areg[d] - sMxOff[p * 32 + d]);
          atomicAdd(&sDenOff[p * 32 + d], e);
          atomicAdd(&sAggOff[p * 32 + d], e * hreg[d]);
        }
      }
    }
  }
  __syncthreads();
  for (int i = tid; i < 1024; i += blockDim.x) {
    atomicAdd(&den_on[i], sDenOn[i]);
    atomicAdd(&den_off[i], sDenOff[i]);
    atomicAdd(&agg_on[i], sAggOn[i]);
    atomicAdd(&agg_off[i], sAggOff[i]);
  }
}

// ---------------------------------------------------------------- final Wm MLP
__global__ void k_final(const float* __restrict__ den_on, const float* __restrict__ den_off,
                        const float* __restrict__ agg_on, const float* __restrict__ agg_off,
                        const float* __restrict__ Wm, const float* __restrict__ bm,
                        float* __restrict__ out, int N) {
  const int p = threadIdx.x;
  if (p >= 32) return;
  for (int s = 0; s < 2; ++s) {
    const float* den = s ? den_off : den_on;
    const float* agg = s ? agg_off : agg_on;
    float a[32];
    #pragma unroll
    for (int d = 0; d < 32; ++d) {
      const float de = den[p * 32 + d];
      a[d] = (de > 0.0f) ? (agg[p * 32 + d] / fmaxf(de, 1e-12f)) : 0.0f;
    }
    const size_t base = (size_t)N * 32 + (size_t)s * 1344 + (size_t)p * 42;
    #pragma unroll 4
    for (int i = 0; i < 32; ++i) {
      float v = bm[i];
      for (int d = 0; d < 32; ++d) v += a[d] * Wm[i * 32 + d];
      out[base + i] = fmaxf(v, 0.0f);
    }
  }
}

extern "C" void kernel_launch(void* const* d_in, const int* in_sizes, int n_in,
                              void* d_out, int out_size, void* d_ws, size_t ws_size,
                              hipStream_t stream) {
  const float* xyz = (const float*)d_in[0];
  const float* center = (const float*)d_in[1];
  const float* feature = (const float*)d_in[2];
  const float* feature_geo = (const float*)d_in[3];
  const float* plane_centers = (const float*)d_in[4];
  const float* normals = (const float*)d_in[5];
  const float* xyz_min = (const float*)d_in[6];
  const float* xyz_max = (const float*)d_in[7];
  const int* axes = (const int*)d_in[8];
  const float* W1 = (const float*)d_in[9];  const float* b1 = (const float*)d_in[10];
  const float* W2 = (const float*)d_in[11]; const float* b2 = (const float*)d_in[12];
  const float* W3 = (const float*)d_in[13]; const float* b3 = (const float*)d_in[14];
  const float* W4 = (const float*)d_in[15]; const float* b4 = (const float*)d_in[16];
  const float* Watt = (const float*)d_in[17];
  const float* Wm = (const float*)d_in[18]; const float* bm = (const float*)d_in[19];
  float* out = (float*)d_out;
  const int N = in_sizes[0] / 3;

  char* wp = (char*)d_ws;
  float* h_ws = (float*)wp;        wp += (size_t)32 * N * 4;
  float* att_ws = (float*)wp;      wp += (size_t)32 * N * 4;
  unsigned* maskb = (unsigned*)wp; wp += (size_t)N * 4;
  unsigned* onb = (unsigned*)wp;   wp += (size_t)N * 4;
  unsigned* offb = (unsigned*)wp;  wp += (size_t)N * 4;
  unsigned* cnt = (unsigned*)wp;   wp += 32 * 4;
  float* mp = (float*)wp;          wp += 1024 * 4;
  float* npws = (float*)wp;        wp += 128 * 4;
  unsigned* vbws = (unsigned*)wp;  wp += 64;
  float* mx_on = (float*)wp;       wp += 4096;
  float* mx_off = (float*)wp;      wp += 4096;
  float* den_on = (float*)wp;      wp += 4096;
  float* den_off = (float*)wp;     wp += 4096;
  float* agg_on = (float*)wp;      wp += 4096;
  float* agg_off = (float*)wp;     wp += 4096;

  k_init<<<4, 256, 0, stream>>>(cnt, mp);

  const int ntiles = (N + 31) / 32;
  int gA = (ntiles + WPB - 1) / WPB;
  if (gA > 1024) gA = 1024;
  if (gA < 1) gA = 1;
  k_phase1<<<gA, WPB * 32, 0, stream>>>(xyz, center, feature, feature_geo,
      plane_centers, normals, xyz_min, xyz_max, axes, W1, b1, W2, b2, Watt,
      h_ws, att_ws, maskb, cnt, mp, N);

  k_planes<<<1, 256, 0, stream>>>(plane_centers, normals, xyz_min, xyz_max,
      W3, b3, W4, b4, cnt, mp, npws, vbws, mx_on, mx_off, den_on, den_off,
      agg_on, agg_off, out, N);

  const int gC = (N + 255) / 256;
  k_project<<<gC, 256, 0, stream>>>(xyz, center, npws, vbws, maskb, att_ws,
      onb, offb, mx_on, mx_off, out, N);

  k_pool<<<gC, 256, 0, stream>>>(onb, offb, att_ws, h_ws, mx_on, mx_off,
      den_on, den_off, agg_on, agg_off, N);

  k_final<<<1, 32, 0, stream>>>(den_on, den_off, agg_on, agg_off, Wm, bm, out, N);
}